// GNN_31069793419526
// MI455X (gfx1250) — compile-verified
//
#include <hip/hip_runtime.h>
#include <hip/hip_bf16.h>

// Problem constants (B, N, D, K) = (4, 384, 64, 2), fp32 throughout.
#define B_ 4
#define N_ 384
#define D_ 64
// X time-step stride (K+1)
#define TS 3

typedef __attribute__((ext_vector_type(2))) float v2f;
typedef __attribute__((ext_vector_type(8))) float v8f;

// ---------------------------------------------------------------------------
// Stage 1: T[b] = X1[b] + A1[b] @ X2[b]        (per batch: 384x384 @ 384x64)
// Grid: B * N/16 blocks, 128 threads (4 waves). Wave w owns cols 16w..16w+15.
// ---------------------------------------------------------------------------
__global__ __launch_bounds__(128) void gnn_stage1(const float* __restrict__ A,
                                                  const float* __restrict__ X,
                                                  float* __restrict__ T) {
    const int wave = threadIdx.x >> 5;      // 0..3
    const int lane = threadIdx.x & 31;
    const int b    = blockIdx.x / (N_ / 16);
    const int i0   = (blockIdx.x % (N_ / 16)) * 16;
    const int d0   = wave * 16;

    const int lm = lane & 15;               // M (A-frag) / N (B-frag)
    const int lk = (lane >> 4) * 2;         // K sub-offset per lane half

    // A[b,i,j,hop] flat: ((b*N+i)*N + j)*2 + hop
    const float* Arow = A + (size_t)(b * N_ + i0 + lm) * N_ * 2;
    const float* Xb   = X + (size_t)b * N_ * D_ * TS;   // X[b,n,d,t] = Xb[(n*D+d)*TS + t]

    v8f c = {};
    for (int k0 = 0; k0 < N_; k0 += 4) {
        const int j = k0 + lk;
        // One b128 load: (A0[j], A1[j], A0[j+1], A1[j+1]) — 16B aligned (j even, N even)
        const float4 af = *(const float4*)(Arow + (size_t)j * 2);
        v2f a;  a.x = af.y;  a.y = af.w;                 // hop 1
        v2f bv;                                          // B frag from X2 (t = 2)
        bv.x = Xb[((size_t)j       * D_ + d0 + lm) * TS + 2];
        bv.y = Xb[((size_t)(j + 1) * D_ + d0 + lm) * TS + 2];
        c = __builtin_amdgcn_wmma_f32_16x16x4_f32(false, a, false, bv,
                                                  (short)0, c, false, false);
    }

    // Epilogue: T = C + X1 (t = 1); C layout: vgpr r -> M = r + 8*(lane/16), N = lm
    const int row0 = (lane >> 4) * 8;
    float* Trow = T + (size_t)(b * N_ + i0) * D_;
#pragma unroll
    for (int r = 0; r < 8; ++r) {
        const int row = row0 + r;
        const int col = d0 + lm;
        const float x1 = Xb[((size_t)(i0 + row) * D_ + col) * TS + 1];
        Trow[(size_t)row * D_ + col] = c[r] + x1;
    }
}

// ---------------------------------------------------------------------------
// Stage 2: out[b] = (X0[b] + A0[b] @ T[b]) @ W
// Same tiling; S (16x64) staged through LDS to re-shape C-layout -> A-layout
// for the fused 64x64 readout GEMM.
// ---------------------------------------------------------------------------
__global__ __launch_bounds__(128) void gnn_stage2(const float* __restrict__ A,
                                                  const float* __restrict__ X,
                                                  const float* __restrict__ T,
                                                  const float* __restrict__ W,
                                                  float* __restrict__ out) {
    __shared__ float S[16][D_ + 4];         // +4 pad: kill bank conflicts

    const int wave = threadIdx.x >> 5;
    const int lane = threadIdx.x & 31;
    const int b    = blockIdx.x / (N_ / 16);
    const int i0   = (blockIdx.x % (N_ / 16)) * 16;
    const int d0   = wave * 16;
    const int lm   = lane & 15;
    const int lk   = (lane >> 4) * 2;
    const int row0 = (lane >> 4) * 8;

    const float* Arow = A + (size_t)(b * N_ + i0 + lm) * N_ * 2;
    const float* Xb   = X + (size_t)b * N_ * D_ * TS;
    const float* Tb   = T + (size_t)b * N_ * D_;

    // S_tile = X0_tile + A0_rowstrip @ T      (contraction over 384)
    v8f c = {};
    for (int k0 = 0; k0 < N_; k0 += 4) {
        const int j = k0 + lk;
        const float4 af = *(const float4*)(Arow + (size_t)j * 2);
        v2f a;  a.x = af.x;  a.y = af.z;                 // hop 0
        v2f bv;
        bv.x = Tb[(size_t)j       * D_ + d0 + lm];
        bv.y = Tb[(size_t)(j + 1) * D_ + d0 + lm];
        c = __builtin_amdgcn_wmma_f32_16x16x4_f32(false, a, false, bv,
                                                  (short)0, c, false, false);
    }
#pragma unroll
    for (int r = 0; r < 8; ++r) {
        const int row = row0 + r;
        const int col = d0 + lm;
        S[row][col] = c[r] + Xb[((size_t)(i0 + row) * D_ + col) * TS + 0];
    }
    __syncthreads();

    // out_tile = S @ W   (16x64 @ 64x64, this wave -> output cols d0..d0+15)
    v8f c2 = {};
    for (int k0 = 0; k0 < D_; k0 += 4) {
        const int k = k0 + lk;
        v2f a;  a.x = S[lm][k];  a.y = S[lm][k + 1];     // A-frag from LDS
        v2f bv;                                          // W row-major (d_in, d_out)
        bv.x = W[(size_t)k       * D_ + d0 + lm];
        bv.y = W[(size_t)(k + 1) * D_ + d0 + lm];
        c2 = __builtin_amdgcn_wmma_f32_16x16x4_f32(false, a, false, bv,
                                                   (short)0, c2, false, false);
    }

    float* Orow = out + (size_t)(b * N_ + i0) * D_;
#pragma unroll
    for (int r = 0; r < 8; ++r) {
        Orow[(size_t)(row0 + r) * D_ + d0 + lm] = c2[r];
    }
}

// ---------------------------------------------------------------------------
extern "C" void kernel_launch(void* const* d_in, const int* in_sizes, int n_in,
                              void* d_out, int out_size, void* d_ws, size_t ws_size,
                              hipStream_t stream) {
    (void)in_sizes; (void)n_in; (void)out_size; (void)ws_size;
    const float* A = (const float*)d_in[0];   // (B, N, N, K)   fp32
    const float* X = (const float*)d_in[1];   // (B, N, D, K+1) fp32
    const float* W = (const float*)d_in[2];   // (D, D)         fp32
    float* out = (float*)d_out;               // (B, N, D)      fp32
    float* T   = (float*)d_ws;                // workspace: B*N*D fp32 = 384 KB

    const dim3 grid(B_ * (N_ / 16));          // 96 blocks
    const dim3 block(128);                    // 4 waves, one 16-col slice each

    gnn_stage1<<<grid, block, 0, stream>>>(A, X, T);
    gnn_stage2<<<grid, block, 0, stream>>>(A, X, T, W, out);
}